// HeterogeneousNetwork_10058813407385
// MI455X (gfx1250) — compile-verified
//
#include <hip/hip_runtime.h>

typedef __attribute__((ext_vector_type(2))) float v2f;
typedef __attribute__((ext_vector_type(8))) float v8f;

#define NROOMS 50000
#define NOBJ   200000
#define DD     64
#define OUTD   25

// ---------------------------------------------------------------------------
// Degree counting (dst histogram) and inversion: deg -> 1/max(deg,1)
// ---------------------------------------------------------------------------
__global__ void deg_count_kernel(const int* __restrict__ dst, float* __restrict__ deg, int n) {
    int i = blockIdx.x * blockDim.x + threadIdx.x;
    if (i < n) unsafeAtomicAdd(&deg[dst[i]], 1.0f);
}

__global__ void inv_deg_kernel(float* __restrict__ deg, int n) {
    int i = blockIdx.x * blockDim.x + threadIdx.x;
    if (i < n) deg[i] = 1.0f / fmaxf(deg[i], 1.0f);
}

// ---------------------------------------------------------------------------
// Edge scatter: one wave per edge; lane handles 2 of 64 features.
// Coalesced 256B row gather + 2 HW float-atomic instructions per edge.
// ---------------------------------------------------------------------------
__global__ __launch_bounds__(256)
void scatter_kernel(const float* __restrict__ x, const int* __restrict__ src,
                    const int* __restrict__ dst, float* __restrict__ agg, int n_edges) {
    int wave = (blockIdx.x * blockDim.x + threadIdx.x) >> 5;
    int lane = threadIdx.x & 31;
    if (wave >= n_edges) return;
    int s = src[wave];
    int d = dst[wave];
    const float2 v = *(const float2*)(x + (size_t)s * DD + lane * 2);
    float* a = agg + (size_t)d * DD + lane * 2;
    unsafeAtomicAdd(a + 0, v.x);
    unsafeAtomicAdd(a + 1, v.y);
}

// ---------------------------------------------------------------------------
// WMMA helper: f32 16x16x4.  A: 16x4 (lanes 0-15 K={0,1}; lanes 16-31 K={2,3})
// B: 4x16 mirrored.  C/D: 8 VGPRs, VGPR v = rows {v, v+8}.
// ---------------------------------------------------------------------------
__device__ inline v8f wmma_f32(v2f a, v2f b, v8f c) {
    return __builtin_amdgcn_wmma_f32_16x16x4_f32(false, a, false, b, (short)0, c, false, false);
}

#define AST 68  // padded LDS row stride (floats) for A tiles: kills bank conflicts

// Weights in LDS are K-pair interleaved: element (k, c) lives at
//   sW[(k>>1)*2*NC + c*2 + (k&1)]  ==  sW[k*NC + ((k&1) ? (c*2+1 - NC) : c*2)]
// so a lane's (W[k][c], W[k+1][c]) B-fragment is one aligned 8-byte ds_load_b64.

// ---------------------------------------------------------------------------
// Fused hidden-layer node update (one dst node type, two incoming edge types):
//   out = relu( mean0@Wl0 + mean1@Wl1 + xd@(Wr0+Wr1) + (b0+b1) )
// Block = 128 threads = 4 waves; each wave owns a 16-row tile (16x64 output).
// ---------------------------------------------------------------------------
__global__ __launch_bounds__(128)
void node_hidden_kernel(const float* __restrict__ agg0, const float* __restrict__ invd0,
                        const float* __restrict__ agg1, const float* __restrict__ invd1,
                        const float* __restrict__ xd,
                        const float* __restrict__ Wl0, const float* __restrict__ Wl1,
                        const float* __restrict__ Wr0, const float* __restrict__ Wr1,
                        const float* __restrict__ b0, const float* __restrict__ b1,
                        float* __restrict__ out, int n_nodes) {
    __shared__ float sW[3][DD * DD];     // Wl0, Wl1, Wr0+Wr1 (K-pair interleaved)
    __shared__ float sBias[DD];
    __shared__ float sA[4][16 * AST];    // per-wave staged A tile

    const int tid = threadIdx.x;
    for (int i = tid; i < DD * DD; i += 128) {
        int k = i >> 6, c = i & 63;
        int di = (k >> 1) * (2 * DD) + c * 2 + (k & 1);   // interleaved index
        sW[0][di] = Wl0[i];
        sW[1][di] = Wl1[i];
        sW[2][di] = Wr0[i] + Wr1[i];
    }
    if (tid < DD) sBias[tid] = b0[tid] + b1[tid];
    __syncthreads();

    const int w = tid >> 5;
    const int lane = tid & 31;
    const int rowbase = (blockIdx.x * 4 + w) * 16;
    if (rowbase >= n_nodes) return;

    v8f acc[4] = {};
    float* myA = sA[w];

    const int m = lane & 15;
    const int kk = (lane >> 4) * 2;

    for (int p = 0; p < 3; ++p) {
        const float* base = (p == 0) ? agg0 : (p == 1) ? agg1 : xd;
        const float* invd = (p == 0) ? invd0 : invd1;
        // stage 16x64 tile (deg-scaled for mean parts), coalesced float4 loads
        for (int i = lane; i < 256; i += 32) {
            int r = i >> 4;
            int c4 = (i & 15) * 4;
            float4 v = *(const float4*)(base + (size_t)(rowbase + r) * DD + c4);
            float sc = (p < 2) ? invd[rowbase + r] : 1.0f;
            float* dp = myA + r * AST + c4;
            dp[0] = v.x * sc; dp[1] = v.y * sc; dp[2] = v.z * sc; dp[3] = v.w * sc;
        }
        asm volatile("s_wait_dscnt 0" ::: "memory");

        const float* wp = sW[p];
        for (int k0 = 0; k0 < DD; k0 += 4) {
            // A fragment: contiguous 8B, aligned (AST even, k0+kk even)
            v2f a = *(const v2f*)&myA[m * AST + k0 + kk];
#pragma unroll
            for (int ct = 0; ct < 4; ++ct) {
                // B fragment: one aligned 8B LDS load ((k0+kk)>>1)*128 == (k0+kk)*64
                v2f b = *(const v2f*)&wp[(k0 + kk) * DD + (ct * 16 + m) * 2];
                acc[ct] = wmma_f32(a, b, acc[ct]);
            }
        }
        // no trailing wait needed: per-wave LDS ops are in-order, so the next
        // part's staging stores cannot pass the fragment loads issued above.
    }

    const int rh = (lane >> 4) * 8;
#pragma unroll
    for (int ct = 0; ct < 4; ++ct) {
        int col = ct * 16 + m;
        float bias = sBias[col];
#pragma unroll
        for (int v = 0; v < 8; ++v) {
            int row = rowbase + rh + v;
            out[(size_t)row * DD + col] = fmaxf(acc[ct][v] + bias, 0.0f);
        }
    }
}

// ---------------------------------------------------------------------------
// Final layer for rooms: 64 -> 25 (padded to 32 cols), no ReLU.
// ---------------------------------------------------------------------------
__global__ __launch_bounds__(128)
void node_out_kernel(const float* __restrict__ agg0, const float* __restrict__ invd0,
                     const float* __restrict__ agg1, const float* __restrict__ invd1,
                     const float* __restrict__ xd,
                     const float* __restrict__ Wl0, const float* __restrict__ Wl1,
                     const float* __restrict__ Wr0, const float* __restrict__ Wr1,
                     const float* __restrict__ b0, const float* __restrict__ b1,
                     float* __restrict__ out, int n_nodes) {
    __shared__ float sW[3][DD * 32];     // K-pair interleaved, 32 padded cols
    __shared__ float sBias[32];
    __shared__ float sA[4][16 * AST];

    const int tid = threadIdx.x;
    for (int i = tid; i < DD * 32; i += 128) {
        int k = i >> 5, c = i & 31;
        bool ok = (c < OUTD);
        int di = (k >> 1) * 64 + c * 2 + (k & 1);         // interleaved index
        sW[0][di] = ok ? Wl0[k * OUTD + c] : 0.0f;
        sW[1][di] = ok ? Wl1[k * OUTD + c] : 0.0f;
        sW[2][di] = ok ? (Wr0[k * OUTD + c] + Wr1[k * OUTD + c]) : 0.0f;
    }
    if (tid < 32) sBias[tid] = (tid < OUTD) ? (b0[tid] + b1[tid]) : 0.0f;
    __syncthreads();

    const int w = tid >> 5;
    const int lane = tid & 31;
    const int rowbase = (blockIdx.x * 4 + w) * 16;
    if (rowbase >= n_nodes) return;

    v8f acc[2] = {};
    float* myA = sA[w];
    const int m = lane & 15;
    const int kk = (lane >> 4) * 2;

    for (int p = 0; p < 3; ++p) {
        const float* base = (p == 0) ? agg0 : (p == 1) ? agg1 : xd;
        const float* invd = (p == 0) ? invd0 : invd1;
        for (int i = lane; i < 256; i += 32) {
            int r = i >> 4;
            int c4 = (i & 15) * 4;
            float4 v = *(const float4*)(base + (size_t)(rowbase + r) * DD + c4);
            float sc = (p < 2) ? invd[rowbase + r] : 1.0f;
            float* dp = myA + r * AST + c4;
            dp[0] = v.x * sc; dp[1] = v.y * sc; dp[2] = v.z * sc; dp[3] = v.w * sc;
        }
        asm volatile("s_wait_dscnt 0" ::: "memory");

        const float* wp = sW[p];
        for (int k0 = 0; k0 < DD; k0 += 4) {
            v2f a = *(const v2f*)&myA[m * AST + k0 + kk];
#pragma unroll
            for (int ct = 0; ct < 2; ++ct) {
                // ((k0+kk)>>1)*64 == (k0+kk)*32
                v2f b = *(const v2f*)&wp[(k0 + kk) * 32 + (ct * 16 + m) * 2];
                acc[ct] = wmma_f32(a, b, acc[ct]);
            }
        }
    }

    const int rh = (lane >> 4) * 8;
#pragma unroll
    for (int ct = 0; ct < 2; ++ct) {
        int col = ct * 16 + m;
        if (col >= OUTD) continue;
        float bias = sBias[col];
#pragma unroll
        for (int v = 0; v < 8; ++v) {
            int row = rowbase + rh + v;
            out[(size_t)row * OUTD + col] = acc[ct][v] + bias;
        }
    }
}

// ---------------------------------------------------------------------------
extern "C" void kernel_launch(void* const* d_in, const int* in_sizes, int n_in,
                              void* d_out, int out_size, void* d_ws, size_t ws_size,
                              hipStream_t stream) {
    const float* x_rooms = (const float*)d_in[0];
    const float* x_obj   = (const float*)d_in[1];
    const float* Wl_hid  = (const float*)d_in[2];   // [2][4][64][64]
    const float* Wr_hid  = (const float*)d_in[3];   // [2][4][64][64]
    const float* b_hid   = (const float*)d_in[4];   // [2][4][64]
    const float* Wl_out  = (const float*)d_in[5];   // [4][64][25]
    const float* Wr_out  = (const float*)d_in[6];   // [4][64][25]
    const float* b_out   = (const float*)d_in[7];   // [4][25]
    const int* src_rr = (const int*)d_in[8];  const int E_rr = in_sizes[8];
    const int* dst_rr = (const int*)d_in[9];
    const int* src_or = (const int*)d_in[10]; const int E_or = in_sizes[10];
    const int* dst_or = (const int*)d_in[11];
    const int* src_ro = (const int*)d_in[12]; const int E_ro = in_sizes[12];
    const int* dst_ro = (const int*)d_in[13];
    const int* src_oo = (const int*)d_in[14]; const int E_oo = in_sizes[14];
    const int* dst_oo = (const int*)d_in[15];

    float* ws = (float*)d_ws;
    size_t off = 0;
    auto alloc = [&](size_t n) { float* p = ws + off; off += n; return p; };
    float* xr1   = alloc((size_t)NROOMS * DD);
    float* xr2   = alloc((size_t)NROOMS * DD);
    float* xo1   = alloc((size_t)NOBJ * DD);
    float* xo2   = alloc((size_t)NOBJ * DD);
    float* aggR0 = alloc((size_t)NROOMS * DD);   // agg buffers contiguous
    float* aggR1 = alloc((size_t)NROOMS * DD);
    float* aggO0 = alloc((size_t)NOBJ * DD);
    float* aggO1 = alloc((size_t)NOBJ * DD);
    float* degR0 = alloc(NROOMS);                // deg buffers contiguous
    float* degR1 = alloc(NROOMS);
    float* degO0 = alloc(NOBJ);
    float* degO1 = alloc(NOBJ);

    // --- degrees: count once (dst sets are layer-invariant), then invert ---
    const int NDEG = 2 * NROOMS + 2 * NOBJ;
    hipMemsetAsync(degR0, 0, (size_t)NDEG * sizeof(float), stream);
    deg_count_kernel<<<(E_rr + 255) / 256, 256, 0, stream>>>(dst_rr, degR0, E_rr);
    deg_count_kernel<<<(E_or + 255) / 256, 256, 0, stream>>>(dst_or, degR1, E_or);
    deg_count_kernel<<<(E_ro + 255) / 256, 256, 0, stream>>>(dst_ro, degO0, E_ro);
    deg_count_kernel<<<(E_oo + 255) / 256, 256, 0, stream>>>(dst_oo, degO1, E_oo);
    inv_deg_kernel<<<(NDEG + 255) / 256, 256, 0, stream>>>(degR0, NDEG);

    const size_t aggAllBytes = ((size_t)2 * NROOMS + 2 * NOBJ) * DD * sizeof(float);
    const int tilesR = NROOMS / 16, tilesO = NOBJ / 16;

    auto run_layer = [&](const float* xr_in, const float* xo_in,
                         float* xr_out, float* xo_out, int l) {
        hipMemsetAsync(aggR0, 0, aggAllBytes, stream);
        scatter_kernel<<<((size_t)E_rr * 32 + 255) / 256, 256, 0, stream>>>(xr_in, src_rr, dst_rr, aggR0, E_rr);
        scatter_kernel<<<((size_t)E_or * 32 + 255) / 256, 256, 0, stream>>>(xo_in, src_or, dst_or, aggR1, E_or);
        scatter_kernel<<<((size_t)E_ro * 32 + 255) / 256, 256, 0, stream>>>(xr_in, src_ro, dst_ro, aggO0, E_ro);
        scatter_kernel<<<((size_t)E_oo * 32 + 255) / 256, 256, 0, stream>>>(xo_in, src_oo, dst_oo, aggO1, E_oo);
        const float* Wl = Wl_hid + (size_t)l * 4 * DD * DD;
        const float* Wr = Wr_hid + (size_t)l * 4 * DD * DD;
        const float* bb = b_hid + (size_t)l * 4 * DD;
        node_hidden_kernel<<<(tilesR + 3) / 4, 128, 0, stream>>>(
            aggR0, degR0, aggR1, degR1, xr_in,
            Wl + 0 * DD * DD, Wl + 1 * DD * DD, Wr + 0 * DD * DD, Wr + 1 * DD * DD,
            bb + 0 * DD, bb + 1 * DD, xr_out, NROOMS);
        node_hidden_kernel<<<(tilesO + 3) / 4, 128, 0, stream>>>(
            aggO0, degO0, aggO1, degO1, xo_in,
            Wl + 2 * DD * DD, Wl + 3 * DD * DD, Wr + 2 * DD * DD, Wr + 3 * DD * DD,
            bb + 2 * DD, bb + 3 * DD, xo_out, NOBJ);
    };

    run_layer(x_rooms, x_obj, xr1, xo1, 0);
    run_layer(xr1, xo1, xr2, xo2, 1);

    // --- final layer: rooms only (edge types rr, or) ---
    hipMemsetAsync(aggR0, 0, (size_t)2 * NROOMS * DD * sizeof(float), stream);
    scatter_kernel<<<((size_t)E_rr * 32 + 255) / 256, 256, 0, stream>>>(xr2, src_rr, dst_rr, aggR0, E_rr);
    scatter_kernel<<<((size_t)E_or * 32 + 255) / 256, 256, 0, stream>>>(xo2, src_or, dst_or, aggR1, E_or);
    node_out_kernel<<<(tilesR + 3) / 4, 128, 0, stream>>>(
        aggR0, degR0, aggR1, degR1, xr2,
        Wl_out + 0 * DD * OUTD, Wl_out + 1 * DD * OUTD,
        Wr_out + 0 * DD * OUTD, Wr_out + 1 * DD * OUTD,
        b_out + 0 * OUTD, b_out + 1 * OUTD, (float*)d_out, NROOMS);
}